// SpatialAttention_11871289606580
// MI455X (gfx1250) — compile-verified
//
#include <hip/hip_runtime.h>

typedef __attribute__((ext_vector_type(2))) float v2f;
typedef __attribute__((ext_vector_type(8))) float v8f;

#define CDIM 64
#define NN   1024
#define BB   4

// Hardware-tanh requirement, checked ONLY in the device pass (the host pass
// parses device code too but has no AMDGCN builtins — that's what tripped
// last round). If the device pass compiles, every FAST_TANH is a single
// v_tanh_f32 on the trans pipe.
#if defined(__HIP_DEVICE_COMPILE__)
#  if __has_builtin(__builtin_amdgcn_tanhf)
#    define FAST_TANH(x) __builtin_amdgcn_tanhf(x)
#  else
#    error "device pass: __builtin_amdgcn_tanhf not available on gfx1250"
#  endif
#else
#  define FAST_TANH(x) tanhf(x) /* host pass: parsed, never executed */
#endif

// ---------------------------------------------------------------------------
// Kernel A: att[b,i,j] = sum_c w3[c]*tanh(x1[b,i,c]+x2[b,j,c]); softmax over j.
// One 256-thread block per (b,i) row; each thread owns 4 j values, processed
// as 4 interleaved accumulator chains (tanh on trans pipe co-executes with
// the fma stream; 4-way ILP hides trans latency). x2 (256 KB/batch) is
// L2-resident; x1 row + w3 broadcast from LDS.
// ---------------------------------------------------------------------------
__global__ __launch_bounds__(256) void attn_tanh_softmax(const float* __restrict__ x1,
                                                         const float* __restrict__ x2,
                                                         const float* __restrict__ w3,
                                                         float* __restrict__ out) {
  __shared__ float x1row[CDIM];
  __shared__ float w3s[CDIM];
  __shared__ float red[256];

  const int t  = threadIdx.x;
  const int bi = blockIdx.x;        // b*1024 + i
  const int b  = bi >> 10;

  if (t < CDIM) {
    x1row[t] = x1[(size_t)bi * CDIM + t];
    w3s[t]   = w3[t];
  }
  __syncthreads();

  const float4* p0 = (const float4*)(x2 + (size_t)(b * NN + t) * CDIM);
  const float4* p1 = (const float4*)(x2 + (size_t)(b * NN + t + 256) * CDIM);
  const float4* p2 = (const float4*)(x2 + (size_t)(b * NN + t + 512) * CDIM);
  const float4* p3 = (const float4*)(x2 + (size_t)(b * NN + t + 768) * CDIM);

  float a0 = 0.f, a1 = 0.f, a2 = 0.f, a3 = 0.f;
#pragma unroll
  for (int c4 = 0; c4 < CDIM / 4; ++c4) {
    const float4 v0 = p0[c4];
    const float4 v1 = p1[c4];
    const float4 v2 = p2[c4];
    const float4 v3 = p3[c4];
    const int c = c4 * 4;
#pragma unroll
    for (int k = 0; k < 4; ++k) {
      const float w  = w3s[c + k];
      const float xr = x1row[c + k];
      const float e0 = (k == 0) ? v0.x : (k == 1) ? v0.y : (k == 2) ? v0.z : v0.w;
      const float e1 = (k == 0) ? v1.x : (k == 1) ? v1.y : (k == 2) ? v1.z : v1.w;
      const float e2 = (k == 0) ? v2.x : (k == 1) ? v2.y : (k == 2) ? v2.z : v2.w;
      const float e3 = (k == 0) ? v3.x : (k == 1) ? v3.y : (k == 2) ? v3.z : v3.w;
      a0 = fmaf(w, FAST_TANH(xr + e0), a0);
      a1 = fmaf(w, FAST_TANH(xr + e1), a1);
      a2 = fmaf(w, FAST_TANH(xr + e2), a2);
      a3 = fmaf(w, FAST_TANH(xr + e3), a3);
    }
  }

  // row max
  red[t] = fmaxf(fmaxf(a0, a1), fmaxf(a2, a3));
  __syncthreads();
#pragma unroll
  for (int s = 128; s > 0; s >>= 1) {
    if (t < s) red[t] = fmaxf(red[t], red[t + s]);
    __syncthreads();
  }
  const float rowmax = red[0];
  __syncthreads();

  // exp + row sum
  const float e0 = __expf(a0 - rowmax);
  const float e1 = __expf(a1 - rowmax);
  const float e2 = __expf(a2 - rowmax);
  const float e3 = __expf(a3 - rowmax);
  red[t] = (e0 + e1) + (e2 + e3);
  __syncthreads();
#pragma unroll
  for (int s = 128; s > 0; s >>= 1) {
    if (t < s) red[t] += red[t + s];
    __syncthreads();
  }
  const float inv = 1.0f / red[0];

  float* orow = out + (size_t)bi * NN;
  orow[t]       = e0 * inv;
  orow[t + 256] = e1 * inv;
  orow[t + 512] = e2 * inv;
  orow[t + 768] = e3 * inv;
}

// ---------------------------------------------------------------------------
// Kernel B: projection GEMMs via V_WMMA_F32_16X16X4_F32 (wave32).
// y[m,d] = sum_c x[m,c] * W[d,c]   (x:[4096,64], W:[64,64] row-major [out,in])
// One wave per 16x16 output tile; K=64 done as 16 WMMA steps of K=4.
//   A (16x4):  lanes 0-15 hold M=lane, {VGPR0,VGPR1}={K0,K1};
//              lanes 16-31 hold M=lane-16, {K2,K3}
//   B (4x16):  VGPR0: rows K0 / K2 in lane halves; VGPR1: rows K1 / K3
//   D (16x16): VGPR r: lanes0-15 -> M=r, N=lane; lanes16-31 -> M=r+8
// ---------------------------------------------------------------------------
__global__ __launch_bounds__(32) void proj_wmma_f32(const float* __restrict__ x,
                                                    const float* __restrict__ W1,
                                                    const float* __restrict__ W2,
                                                    float* __restrict__ y1,
                                                    float* __restrict__ y2) {
  const int lane = threadIdx.x & 31;
  const int rc   = lane & 15;   // A row index == B col index for this lane
  const int kh   = lane >> 4;   // which K-pair half
  const int m0   = blockIdx.x * 16;
  const int d0   = blockIdx.y * 16;
  const float* __restrict__ W = (blockIdx.z == 0) ? W1 : W2;
  float* __restrict__ y       = (blockIdx.z == 0) ? y1 : y2;

  const float* xrow = x + (size_t)(m0 + rc) * CDIM + 2 * kh;
  const float* wrow = W + (size_t)(d0 + rc) * CDIM + 2 * kh;

  v8f c = {};
#pragma unroll
  for (int ks = 0; ks < 16; ++ks) {
    v2f a = *(const v2f*)(xrow + 4 * ks);  // 8B-aligned: offset is even
    v2f b = *(const v2f*)(wrow + 4 * ks);
    c = __builtin_amdgcn_wmma_f32_16x16x4_f32(false, a, false, b, (short)0, c,
                                              false, false);
  }

  float* outp = y + (size_t)(m0 + 8 * kh) * CDIM + d0 + rc;
#pragma unroll
  for (int r = 0; r < 8; ++r) outp[r * CDIM] = c[r];
}

extern "C" void kernel_launch(void* const* d_in, const int* in_sizes, int n_in,
                              void* d_out, int out_size, void* d_ws, size_t ws_size,
                              hipStream_t stream) {
  (void)in_sizes; (void)n_in; (void)out_size; (void)ws_size;
  const float* x  = (const float*)d_in[0];  // [4,1024,64]
  const float* W1 = (const float*)d_in[1];  // [64,64]
  const float* W2 = (const float*)d_in[2];  // [64,64]
  const float* w3 = (const float*)d_in[3];  // [64]
  float* out = (float*)d_out;               // [4,1024,1024]

  float* x1 = (float*)d_ws;                 // [4096,64]
  float* x2 = x1 + (size_t)BB * NN * CDIM;  // [4096,64]  (2 MB total scratch)

  dim3 g1((BB * NN) / 16, CDIM / 16, 2);    // 256 x 4 x 2 tiles, 1 wave each
  proj_wmma_f32<<<g1, 32, 0, stream>>>(x, W1, W2, x1, x2);

  attn_tanh_softmax<<<BB * NN, 256, 0, stream>>>(x1, x2, w3, out);
}